// BatchTopK_37769942401182
// MI455X (gfx1250) — compile-verified
//
#include <hip/hip_runtime.h>
#include <hip/hip_bf16.h>

// ---------------------------------------------------------------------------
// BatchTopK (training mode): global rank-(B*K) cutoff via 4-pass MSB radix
// select (input stays resident in the 192MB L2 across passes), then a single
// streaming masked-ReLU pass with non-temporal stores. Wave32 / gfx1250.
// Histogram passes stream data via GLOBAL_LOAD_ASYNC_TO_LDS (ASYNCcnt) when
// the toolchain exposes the builtin; otherwise plain b128 loads.
// ---------------------------------------------------------------------------

typedef __attribute__((ext_vector_type(4))) float float4v;

#define WS_PREFIX 0    // selected key prefix (high bits)
#define WS_REMK   1    // remaining rank within current prefix
#define WS_CUTBITS 2   // final cutoff (float bits)
#define WS_BINS   8    // 256 histogram bins start here

#define HIST_COPIES 32
#define HIST_STRIDE 257   // 257 % 64 == 1 -> a hot bin maps the 32 lanes to 32 banks

#if defined(__HIP_DEVICE_COMPILE__) && __has_builtin(__builtin_amdgcn_global_load_async_to_lds_b128)
#define HAVE_ASYNC_LDS 1
#else
#define HAVE_ASYNC_LDS 0
#endif

__device__ __forceinline__ unsigned sort_key(float f) {
    unsigned b = __float_as_uint(f);
    // ascending-order key: flip sign bit for positives, flip all bits for negatives
    return b ^ ((unsigned)((int)b >> 31) | 0x80000000u);
}

#if HAVE_ASYNC_LDS
// builtin signature (from hipcc diagnostic): (v4i AS(1)*, v4i AS(3)*, Ii offset, Ii cpol)
typedef __attribute__((vector_size(16))) int v4i;
typedef __attribute__((address_space(1))) v4i* v4i_gptr;
typedef __attribute__((address_space(3))) v4i* v4i_lptr;

__device__ __forceinline__ void async_ld_b128(const float4v* g, float4v* l) {
    __builtin_amdgcn_global_load_async_to_lds_b128(
        (v4i_gptr)(v4i*)(void*)g,
        (v4i_lptr)(v4i*)(void*)l,
        0, 0);
}
#endif

__global__ void btk_init_kernel(unsigned* ws, unsigned total_k) {
    int tid = threadIdx.x;
    ws[WS_BINS + tid] = 0u;
    if (tid == 0) {
        ws[WS_PREFIX]  = 0u;
        ws[WS_REMK]    = total_k;
        ws[WS_CUTBITS] = 0u;
    }
}

template <int SHIFT>
__global__ void btk_hist_kernel(const float4v* __restrict__ x4, unsigned n4,
                                unsigned* __restrict__ ws) {
    __shared__ unsigned h[HIST_COPIES * HIST_STRIDE];
#if HAVE_ASYNC_LDS
    __shared__ float4v stage[2][256];   // 16B/lane double buffer
#endif
    const int tid = threadIdx.x;
    for (int i = tid; i < HIST_COPIES * HIST_STRIDE; i += blockDim.x) h[i] = 0u;

    const unsigned prefix = ws[WS_PREFIX];   // uniform scalar load
    __syncthreads();

    const unsigned lane_off = (unsigned)(tid & 31) * HIST_STRIDE;
    const unsigned stride = gridDim.x * blockDim.x;
    unsigned i = blockIdx.x * blockDim.x + tid;

#if HAVE_ASYNC_LDS
    // Double-buffered async stream: each lane reads only the LDS slot its own
    // async load filled, so ordering needs only the per-wave ASYNCcnt wait
    // (async loads complete in order) -- no workgroup barriers in the loop.
    int p = 0;
    if (i < n4) async_ld_b128(&x4[i], &stage[0][tid]);
    for (; i < n4; i += stride) {
        const unsigned inext = i + stride;
        if (inext < n4) {
            async_ld_b128(&x4[inext], &stage[p ^ 1][tid]);
            asm volatile("s_wait_asynccnt 0x1" ::: "memory"); // older tile done
        } else {
            asm volatile("s_wait_asynccnt 0x0" ::: "memory"); // last tile done
        }
        float4v v = stage[p][tid];
        p ^= 1;
#pragma unroll
        for (int j = 0; j < 4; ++j) {
            unsigned key = sort_key(v[j]);
            bool ok;
            if constexpr (SHIFT == 24) ok = true;
            else ok = (((key ^ prefix) >> (SHIFT + 8)) == 0u);
            if (ok) atomicAdd(&h[lane_off + ((key >> SHIFT) & 255u)], 1u);
        }
    }
#else
    for (; i < n4; i += stride) {
        __builtin_prefetch(&x4[i + stride], 0, 0);   // global_prefetch_b8
        float4v v = x4[i];
#pragma unroll
        for (int j = 0; j < 4; ++j) {
            unsigned key = sort_key(v[j]);
            bool ok;
            if constexpr (SHIFT == 24) ok = true;
            else ok = (((key ^ prefix) >> (SHIFT + 8)) == 0u);
            if (ok) atomicAdd(&h[lane_off + ((key >> SHIFT) & 255u)], 1u);
        }
    }
#endif
    __syncthreads();

    // merge 32 lane-private copies, then one global atomic per bin
    for (int b = tid; b < 256; b += blockDim.x) {
        unsigned s = 0;
#pragma unroll
        for (int c = 0; c < HIST_COPIES; ++c) s += h[c * HIST_STRIDE + b];
        if (s) atomicAdd(&ws[WS_BINS + b], s);
    }
}

template <int SHIFT>
__global__ void btk_scan_kernel(unsigned* __restrict__ ws,
                                const float* __restrict__ thr,
                                float* __restrict__ out_thr) {
    __shared__ unsigned c[256];
    const int tid = threadIdx.x;
    c[tid] = ws[WS_BINS + tid];
    ws[WS_BINS + tid] = 0u;          // reset bins for the next pass
    __syncthreads();
    if (tid == 0) {
        unsigned remK = ws[WS_REMK];
        unsigned pfx  = ws[WS_PREFIX];
        unsigned cum  = 0;
        int d = 0;
        for (int i = 255; i >= 0; --i) {       // descending rank scan
            unsigned nc = cum + c[i];
            if (nc >= remK) { d = i; break; }
            cum = nc;
        }
        pfx |= ((unsigned)d) << SHIFT;
        ws[WS_PREFIX] = pfx;
        ws[WS_REMK]   = remK - cum;
        if constexpr (SHIFT == 0) {
            // decode exact cutoff float from its sort key
            unsigned bits = (pfx & 0x80000000u) ? (pfx ^ 0x80000000u) : ~pfx;
            ws[WS_CUTBITS] = bits;
            float cutoff = __uint_as_float(bits);
            out_thr[0] = thr[0] * 0.99f + (1.0f - 0.99f) * cutoff;
        }
    }
}

__global__ void btk_out_kernel(const float4v* __restrict__ x4,
                               float4v* __restrict__ o4, unsigned n4,
                               const unsigned* __restrict__ ws) {
    const float cutoff = __uint_as_float(ws[WS_CUTBITS]);
    const unsigned stride = gridDim.x * blockDim.x;
    for (unsigned i = blockIdx.x * blockDim.x + threadIdx.x; i < n4; i += stride) {
        __builtin_prefetch(&x4[i + stride], 0, 0);
        float4v v = x4[i];
        float4v r;
#pragma unroll
        for (int j = 0; j < 4; ++j)
            r[j] = (v[j] >= cutoff) ? fmaxf(v[j], 0.0f) : 0.0f;
        // non-temporal store: keep the L2-resident input from being evicted
        __builtin_nontemporal_store(r, &o4[i]);
    }
}

extern "C" void kernel_launch(void* const* d_in, const int* in_sizes, int n_in,
                              void* d_out, int out_size, void* d_ws, size_t ws_size,
                              hipStream_t stream) {
    const float* x   = (const float*)d_in[0];   // pre_act, (2048, 16384) f32
    const float* thr = (const float*)d_in[1];   // threshold scalar f32
    float* out = (float*)d_out;                  // [N masked output] ++ [1 new_threshold]
    unsigned* ws = (unsigned*)d_ws;

    const unsigned n  = (unsigned)in_sizes[0];   // 33,554,432
    const unsigned n4 = n / 4u;
    const unsigned d_model = 16384u;
    const unsigned total_k = (n / d_model) * 64u; // B * K = 131072

    const float4v* x4 = (const float4v*)x;
    float4v* o4 = (float4v*)out;
    float* out_thr = out + n;

    const dim3 hb(256);
    const dim3 hg(1024);      // 1024 WGs x 256 thr -> 32 float4 per thread

    btk_init_kernel<<<1, 256, 0, stream>>>(ws, total_k);

    btk_hist_kernel<24><<<hg, hb, 0, stream>>>(x4, n4, ws);   // HBM pass
    btk_scan_kernel<24><<<1, 256, 0, stream>>>(ws, thr, out_thr);

    btk_hist_kernel<16><<<hg, hb, 0, stream>>>(x4, n4, ws);   // L2-resident
    btk_scan_kernel<16><<<1, 256, 0, stream>>>(ws, thr, out_thr);

    btk_hist_kernel<8><<<hg, hb, 0, stream>>>(x4, n4, ws);    // L2-resident
    btk_scan_kernel<8><<<1, 256, 0, stream>>>(ws, thr, out_thr);

    btk_hist_kernel<0><<<hg, hb, 0, stream>>>(x4, n4, ws);    // L2-resident
    btk_scan_kernel<0><<<1, 256, 0, stream>>>(ws, thr, out_thr); // writes cutoff + EMA

    btk_out_kernel<<<4096, 256, 0, stream>>>(x4, o4, n4, ws); // stream out, NT stores
}